// GraphNeuralNet_7670811591303
// MI455X (gfx1250) — compile-verified
//
#include <hip/hip_runtime.h>

#define N_NODES  100000
#define N_EDGES  1600000
#define N_GRAPHS 1000
#define FP       112           // padded feature stride (7 x 16)
#define NT       7             // column tiles per wave (7*16 = 112)
#define MP       1008          // pooled rows padded to multiple of 16

typedef __attribute__((ext_vector_type(2))) float v2f;
typedef __attribute__((ext_vector_type(8))) float v8f;

// ---------------- degree / dinv ----------------
__global__ void k_init_deg(float* deg) {
    int i = blockIdx.x * blockDim.x + threadIdx.x;
    if (i < N_NODES) deg[i] = 2.0f;                  // self loop weight 2
}
__global__ void k_count_deg(const int* __restrict__ col, float* deg) {
    int e = blockIdx.x * blockDim.x + threadIdx.x;
    if (e < N_EDGES) atomicAdd(&deg[col[e]], 1.0f);
}
__global__ void k_deg_to_dinv(float* deg) {
    int i = blockIdx.x * blockDim.x + threadIdx.x;
    if (i < N_NODES) deg[i] = rsqrtf(deg[i]);
}

// ---------------- WMMA GEMM:  C[M x ldc](cols<Ncols) = A[M x lda](cols<K) * B[K x Ncols] ----------------
// block = 128 threads (4 waves). Each wave owns a 16-row stripe and all NT=7 col tiles.
// B staged in LDS pair-interleaved: Bs2[q*FP + n] = (B[2q][n], B[2q+1][n]) so a lane's
// WMMA B-fragment is ONE aligned ds_load_b64 into an even VGPR pair (no repack movs).
__global__ void k_gemm_wmma(const float* __restrict__ A, int lda,
                            const float* __restrict__ B, int K, int Ncols,
                            float* __restrict__ C, int ldc,
                            int M, int Mvalid,
                            const float* __restrict__ bias, int relu) {
    __shared__ v2f Bs2[56 * FP];                      // 56 pair-rows x 112 cols x 8B = 50176 B
    const int tid  = threadIdx.x;
    const int wave = tid >> 5;
    const int lane = tid & 31;
    const int hi   = lane >> 4;                       // 0: K=k,k+1   1: K=k+2,k+3
    const int lo   = lane & 15;
    const int rowbase = blockIdx.x * 64 + wave * 16;
    const bool active = rowbase < M;
    const float* arow = A + (size_t)(active ? (rowbase + lo) : lo) * lda + 2 * hi;

    v8f acc[NT] = {};                                 // 7 x 16x16 f32 accumulators

    for (int k0 = 0; k0 < K; k0 += 112) {
        const int kc = (K - k0 < 112) ? (K - k0) : 112;   // 112 or 100 -> even
        const int qp = kc >> 1;                            // pair-rows this chunk
        for (int idx = tid; idx < qp * FP; idx += blockDim.x) {
            int q = idx / FP, nn = idx % FP;
            v2f pr;
            if (nn < Ncols) {
                pr.x = B[(size_t)(k0 + 2 * q) * Ncols + nn];
                pr.y = B[(size_t)(k0 + 2 * q + 1) * Ncols + nn];
            } else {
                pr.x = 0.0f; pr.y = 0.0f;
            }
            Bs2[idx] = pr;
        }
        __syncthreads();
        if (active) {
            for (int k = 0; k < kc; k += 4) {
                // A fragment: contiguous pair -> aligned 8B global load
                v2f a = *(const v2f*)(arow + k0 + k);
                const int bo = ((k >> 1) + hi) * FP + lo; // pair-row (k/2 + hi), col lo
#pragma unroll
                for (int t = 0; t < NT; ++t) {
                    v2f b = Bs2[bo + t * 16];             // single ds_load_b64
                    acc[t] = __builtin_amdgcn_wmma_f32_16x16x4_f32(
                        false, a, false, b, (short)0, acc[t], false, false);
                }
            }
        }
        __syncthreads();
    }
    if (!active) return;

#pragma unroll
    for (int t = 0; t < NT; ++t) {
        const int n = t * 16 + lo;
        if (n >= Ncols) continue;
        const float bv = bias ? bias[n] : 0.0f;
#pragma unroll
        for (int r = 0; r < 8; ++r) {
            const int row = rowbase + r + 8 * hi;     // C/D layout: VGPR r -> M=r (+8 for hi half)
            if (row >= Mvalid) continue;
            float v = acc[t][r] + bv;
            if (relu) v = fmaxf(v, 0.0f);
            C[(size_t)row * ldc + n] = v;
        }
    }
}

// ---------------- self-loop contribution (also initializes agg over poison) ----------------
__global__ void k_self_contrib(const float* __restrict__ xw, const float* __restrict__ dinv,
                               float* __restrict__ agg) {
    const int node = blockIdx.x;
    const int f = threadIdx.x;
    if (f < 100) {
        const float d = dinv[node];
        agg[(size_t)node * FP + f] = 2.0f * d * d * xw[(size_t)node * FP + f];
    }
}

// ---------------- edge scatter: one wave per edge, float4-coalesced ----------------
__global__ void k_scatter_edges(const float* __restrict__ xw,
                                const int* __restrict__ row, const int* __restrict__ col,
                                const float* __restrict__ dinv, float* __restrict__ agg) {
    const int gt = blockIdx.x * blockDim.x + threadIdx.x;
    const int e = gt >> 5, lane = gt & 31;
    if (e >= N_EDGES) return;
    const int r = row[e], c = col[e];
    const float w = dinv[r] * dinv[c];
    if (lane < 25) {                                   // 25 * float4 = 100 floats
        const float4 v = *(const float4*)(xw + (size_t)r * FP + lane * 4);
        float* dst = agg + (size_t)c * FP + lane * 4;
        atomicAdd(dst + 0, w * v.x);
        atomicAdd(dst + 1, w * v.y);
        atomicAdd(dst + 2, w * v.z);
        atomicAdd(dst + 3, w * v.w);
    }
}

__global__ void k_bias_relu(float* __restrict__ h, const float* __restrict__ bias) {
    const int node = blockIdx.x;
    const int f = threadIdx.x;
    if (f < 100) {
        const size_t idx = (size_t)node * FP + f;
        h[idx] = fmaxf(h[idx] + bias[f], 0.0f);
    }
}

// ---------------- pooling ----------------
__global__ void k_zero(float* p, int n) {
    int i = blockIdx.x * blockDim.x + threadIdx.x;
    if (i < n) p[i] = 0.0f;
}
__global__ void k_pool_accum(const float* __restrict__ h, const int* __restrict__ batch,
                             float* __restrict__ sums, float* __restrict__ counts) {
    const int gt = blockIdx.x * blockDim.x + threadIdx.x;
    const int node = gt >> 5, lane = gt & 31;
    if (node >= N_NODES) return;
    const int g = batch[node];
    if (lane < 25) {
        const float4 v = *(const float4*)(h + (size_t)node * FP + lane * 4);
        float* dst = sums + (size_t)g * FP + lane * 4;
        atomicAdd(dst + 0, v.x);
        atomicAdd(dst + 1, v.y);
        atomicAdd(dst + 2, v.z);
        atomicAdd(dst + 3, v.w);
    }
    if (lane == 0) atomicAdd(&counts[g], 1.0f);
}
__global__ void k_pool_finish(const float* __restrict__ sums, const float* __restrict__ counts,
                              float* __restrict__ pooled) {
    const int g = blockIdx.x;                          // grid = MP (incl. zero pad rows)
    const int f = threadIdx.x;
    if (f < 100) {
        float v = 0.0f;
        if (g < N_GRAPHS) v = sums[(size_t)g * FP + f] / fmaxf(counts[g], 1.0f);
        pooled[(size_t)g * FP + f] = v;
    }
}

// ---------------- launch ----------------
extern "C" void kernel_launch(void* const* d_in, const int* in_sizes, int n_in,
                              void* d_out, int out_size, void* d_ws, size_t ws_size,
                              hipStream_t stream) {
    const float* x     = (const float*)d_in[0];       // [N, 336]
    const int*   edge  = (const int*)d_in[1];         // [2, E]
    const int*   batch = (const int*)d_in[2];         // [N]
    const float* W[5]  = {(const float*)d_in[3], (const float*)d_in[5], (const float*)d_in[7],
                          (const float*)d_in[9], (const float*)d_in[11]};
    const float* b[5]  = {(const float*)d_in[4], (const float*)d_in[6], (const float*)d_in[8],
                          (const float*)d_in[10], (const float*)d_in[12]};
    const float* Wl1 = (const float*)d_in[13]; const float* bl1 = (const float*)d_in[14];
    const float* Wl2 = (const float*)d_in[15]; const float* bl2 = (const float*)d_in[16];
    const float* Wl3 = (const float*)d_in[17]; const float* bl3 = (const float*)d_in[18];
    const int* row = edge;
    const int* col = edge + N_EDGES;
    float* out = (float*)d_out;                        // [1000, 29]

    // workspace carve-up (floats)
    float* p = (float*)d_ws;
    float* dinv   = p;                  p += N_NODES;
    float* xw     = p;                  p += (size_t)N_NODES * FP;
    float* hA     = p;                  p += (size_t)N_NODES * FP;
    float* hB     = p;                  p += (size_t)N_NODES * FP;
    float* sums   = p;                  p += (size_t)MP * FP;
    float* counts = p;                  p += N_GRAPHS;   // contiguous with sums for one zero pass
    float* pooled = p;                  p += (size_t)MP * FP;
    float* g1     = p;                  p += (size_t)MP * FP;
    float* g2     = p;                  p += (size_t)MP * FP;

    // degree -> dinv
    k_init_deg   <<<(N_NODES + 255) / 256, 256, 0, stream>>>(dinv);
    k_count_deg  <<<(N_EDGES + 255) / 256, 256, 0, stream>>>(col, dinv);
    k_deg_to_dinv<<<(N_NODES + 255) / 256, 256, 0, stream>>>(dinv);

    const int gemm_grid_nodes = (N_NODES + 63) / 64;
    const int scatter_grid = (int)(((long long)N_EDGES * 32 + 255) / 256);

    const float* hin = x;
    int lda = 336, K = 336;
    float* hbuf[2] = {hA, hB};
    for (int l = 0; l < 5; ++l) {
        float* agg = hbuf[l & 1];
        k_gemm_wmma<<<gemm_grid_nodes, 128, 0, stream>>>(hin, lda, W[l], K, 100,
                                                         xw, FP, N_NODES, N_NODES, nullptr, 0);
        k_self_contrib <<<N_NODES, 128, 0, stream>>>(xw, dinv, agg);
        k_scatter_edges<<<scatter_grid, 256, 0, stream>>>(xw, row, col, dinv, agg);
        k_bias_relu    <<<N_NODES, 128, 0, stream>>>(agg, b[l]);
        hin = agg; lda = FP; K = 100;
    }
    const float* h5 = hbuf[0];                         // layers 1,3,5 -> hA

    // global mean pool
    k_zero<<<((MP * FP + N_GRAPHS) + 255) / 256, 256, 0, stream>>>(sums, MP * FP + N_GRAPHS);
    k_pool_accum <<<(int)(((long long)N_NODES * 32 + 255) / 256), 256, 0, stream>>>(h5, batch, sums, counts);
    k_pool_finish<<<MP, 128, 0, stream>>>(sums, counts, pooled);

    // MLP head
    const int gemm_grid_mlp = (MP + 63) / 64;
    k_gemm_wmma<<<gemm_grid_mlp, 128, 0, stream>>>(pooled, FP, Wl1, 100, 100, g1, FP, MP, MP, bl1, 1);
    k_gemm_wmma<<<gemm_grid_mlp, 128, 0, stream>>>(g1,     FP, Wl2, 100, 100, g2, FP, MP, MP, bl2, 1);
    k_gemm_wmma<<<gemm_grid_mlp, 128, 0, stream>>>(g2,     FP, Wl3, 100,  29, out, 29, MP, N_GRAPHS, bl3, 0);
}